// Reg_version_wave_72481868087389
// MI455X (gfx1250) — compile-verified
//
#include <hip/hip_runtime.h>

// CDNA5 / gfx1250: wave32, WMMA (not MFMA).
typedef __attribute__((ext_vector_type(2))) float v2f;
typedef __attribute__((ext_vector_type(8))) float v8f;

#define NN 512
#define KDIAGS 510     // valid diagonals d = 1..510
#define TILE_D 128     // diagonals per block (8 waves * 16)
#define WAVE_D 16      // diagonals per wave
#define NWAVES 8       // 256 threads / wave32

// One block = 8 waves; wave w handles diagonals [d0, d0+16) with
// d0 = tile*128 + w*16.  Per WMMA step the wave consumes a 16(diag) x 4(row)
// tile of Y[p][d] = SSMap[b][p][p+d] (zero past the matrix edge):
//   A (16x4 f32, documented layout: lane = M, K = v + 2*(lane>=16))
//   B = ones(4x16)  -> all-ones is layout-invariant
//   C += A x B  accumulates S1[d] (and S2[d] for the squared tile) in f32.
__global__ __launch_bounds__(256) void diag_var_wmma_kernel(
    const float* __restrict__ S, float* __restrict__ partial)
{
    const int b    = blockIdx.x >> 2;   // 4 diagonal tiles per batch
    const int tile = blockIdx.x & 3;
    const int wave = threadIdx.x >> 5;
    const int lane = threadIdx.x & 31;
    const int kh   = lane >> 4;         // lane half selects K group {0,1} vs {2,3}
    const int m    = lane & 15;         // A-matrix row index M = diagonal within strip

    // d0 is wave-uniform; readfirstlane makes loop bounds scalar (SALU control).
    const int d0 = __builtin_amdgcn_readfirstlane(tile * TILE_D + wave * WAVE_D);
    const int d  = d0 + m;              // this lane's diagonal index (0..511)

    const float* Sb = S + (size_t)b * NN * NN;

    v8f acc1 = {};                      // S1 accumulator (16x16 f32 C/D)
    v8f acc2 = {};                      // S2 accumulator
    v2f bones; bones.x = 1.0f; bones.y = 1.0f;

    // element address for row r, diagonal d: r*NN + (r+d) = r*(NN+1) + d
    // Bulk region: all lanes in-bounds when worst lane (r = p0+3, d = d0+15)
    // satisfies p0 + 3 + d0 + 15 <= NN-1  ->  p0 < NN - 18 - d0.
    const int bulk_end = NN - 18 - d0;       // scalar; may be <= 0 for last strips
    const int pmax     = NN - 1 - d0;        // scalar; largest valid p (lane m=0)

    int p0 = 0;
    size_t base = (size_t)(2 * kh) * (NN + 1) + d;   // lane's address at p0 = 0

    // ---- bulk: unconditional loads, one-deep software pipeline ----
    if (bulk_end > 0) {
        float a0 = Sb[base];
        float a1 = Sb[base + (NN + 1)];
        for (; p0 + 4 < bulk_end; p0 += 4) {
            base += (size_t)4 * (NN + 1);
            const float n0 = Sb[base];               // prefetch next step
            const float n1 = Sb[base + (NN + 1)];
            v2f A1; A1.x = a0;      A1.y = a1;
            v2f A2; A2.x = a0 * a0; A2.y = a1 * a1;
            acc1 = __builtin_amdgcn_wmma_f32_16x16x4_f32(
                false, A1, false, bones, (short)0, acc1, false, false);
            acc2 = __builtin_amdgcn_wmma_f32_16x16x4_f32(
                false, A2, false, bones, (short)0, acc2, false, false);
            a0 = n0; a1 = n1;
        }
        {   // epilogue: consume the last preloaded pair
            v2f A1; A1.x = a0;      A1.y = a1;
            v2f A2; A2.x = a0 * a0; A2.y = a1 * a1;
            acc1 = __builtin_amdgcn_wmma_f32_16x16x4_f32(
                false, A1, false, bones, (short)0, acc1, false, false);
            acc2 = __builtin_amdgcn_wmma_f32_16x16x4_f32(
                false, A2, false, bones, (short)0, acc2, false, false);
            p0 += 4;
        }
    }

    // ---- tail: <=5 iterations, branch-free guard via clamp + select ----
    for (; p0 <= pmax; p0 += 4) {
        const int r0 = p0 + 2 * kh;
        const int r1 = r0 + 1;
        const bool ok0 = (r0 + d) < NN;
        const bool ok1 = (r1 + d) < NN;
        const float t0 = Sb[ok0 ? ((size_t)r0 * (NN + 1) + d) : (size_t)0];
        const float t1 = Sb[ok1 ? ((size_t)r1 * (NN + 1) + d) : (size_t)0];
        const float a0 = ok0 ? t0 : 0.0f;
        const float a1 = ok1 ? t1 : 0.0f;
        v2f A1; A1.x = a0;      A1.y = a1;
        v2f A2; A2.x = a0 * a0; A2.y = a1 * a1;
        acc1 = __builtin_amdgcn_wmma_f32_16x16x4_f32(
            false, A1, false, bones, (short)0, acc1, false, false);
        acc2 = __builtin_amdgcn_wmma_f32_16x16x4_f32(
            false, A2, false, bones, (short)0, acc2, false, false);
    }

    // C/D layout (documented): VGPR r, lane L -> M = r + 8*(L>>4); all N equal.
    // Each diagonal's sum is replicated across 16 lanes -> divide by 16 later.
    float local = 0.0f;
#pragma unroll
    for (int r = 0; r < 8; ++r) {
        const int dd = d0 + r + 8 * kh;
        if (dd >= 1 && dd <= KDIAGS) {
            const float s1   = acc1[r];
            const float s2   = acc2[r];
            const float len  = (float)(NN - dd);
            const float mean = s1 / len;
            float var = (s2 - s1 * mean) / (len - 1.0f);  // = sum((x-mean)^2)/(len-1)
            var = fmaxf(var, 0.0f);
            local += sqrtf(var) * len * 0.2f;             // sqrt(var) * len / 5
        }
    }

    // wave32 butterfly reduction
#pragma unroll
    for (int off = 16; off >= 1; off >>= 1)
        local += __shfl_xor(local, off, 32);

    __shared__ float wsum[NWAVES];
    if (lane == 0) wsum[wave] = local * (1.0f / 16.0f);   // undo 16x replication
    __syncthreads();
    if (threadIdx.x == 0) {
        float s = 0.0f;
#pragma unroll
        for (int w = 0; w < NWAVES; ++w) s += wsum[w];
        partial[blockIdx.x] = s;
    }
}

__global__ void diag_var_finalize_kernel(const float* __restrict__ partial,
                                         float* __restrict__ out)
{
    const int b = threadIdx.x;
    if (b < 128) {
        const float s = partial[4 * b + 0] + partial[4 * b + 1] +
                        partial[4 * b + 2] + partial[4 * b + 3];
        const float loss = s / (float)KDIAGS;
        out[b]       = loss;   // loss
        out[128 + b] = loss;   // stop_gradient(loss)
    }
}

extern "C" void kernel_launch(void* const* d_in, const int* in_sizes, int n_in,
                              void* d_out, int out_size, void* d_ws, size_t ws_size,
                              hipStream_t stream) {
    (void)in_sizes; (void)n_in; (void)out_size; (void)ws_size;
    const float* S   = (const float*)d_in[0];
    float* out       = (float*)d_out;
    float* partial   = (float*)d_ws;          // needs 512 floats of scratch

    dim3 grid(128 * 4);                       // 128 batches x 4 diagonal tiles
    dim3 block(256);                          // 8 waves (wave32)
    diag_var_wmma_kernel<<<grid, block, 0, stream>>>(S, partial);
    diag_var_finalize_kernel<<<1, 128, 0, stream>>>(partial, out);
}